// EnhancedRGCNEncoder_15831249453466
// MI455X (gfx1250) — compile-verified
//
#include <hip/hip_runtime.h>
#include <hip/hip_bf16.h>

// ---------------------------------------------------------------------------
// RGCN encoder for MI455X (gfx1250, wave32).
//   - transform-after-aggregate: scatter raw h[src] (25.6MB table, L2-resident)
//     into per-(relation,dst) segment sums, then 9 fp32 WMMA GEMMs per layer.
//   - V_WMMA_F32_16X16X4_F32 keeps full fp32 fidelity (kernel is scatter-bound,
//     matrix throughput is not the bottleneck).
//   - LN+ReLU+residual fused in-register on the WMMA C/D layout (shfl_xor
//     reductions within 16-lane halves).
// ---------------------------------------------------------------------------

#define N_NODES 50000
#define E_EDGES 1600000
#define RELS    8
#define DIM     128
#define LAYERS  3
#define EPSV    1e-5f

typedef __attribute__((ext_vector_type(2))) float v2f;
typedef __attribute__((ext_vector_type(8))) float v8f;

// ---------------------------------------------------------------- zero scratch
__global__ __launch_bounds__(256) void rgcn_zero_kernel(float4* __restrict__ p,
                                                        long n4) {
  long i = (long)blockIdx.x * blockDim.x + threadIdx.x;
  long stride = (long)gridDim.x * blockDim.x;
  for (; i < n4; i += stride) p[i] = make_float4(0.f, 0.f, 0.f, 0.f);
}

// ------------------------------------------------------------- h0 = emb[x[i]]
__global__ __launch_bounds__(256) void rgcn_gather_kernel(
    const float4* __restrict__ emb, const int* __restrict__ x,
    float4* __restrict__ h0) {
  long i = (long)blockIdx.x * blockDim.x + threadIdx.x;  // one float4 each
  const long total = (long)N_NODES * (DIM / 4);
  if (i >= total) return;
  long node = i / (DIM / 4);
  long c4 = i - node * (DIM / 4);
  long srcn = x[node];
  h0[node * (DIM / 4) + c4] = emb[srcn * (DIM / 4) + c4];
}

// ---------------------------------------------- per-edge scatter (wave/edge)
// S[(r*N + dst)*D + :] += h[src*D + :] ; cnt[r*N + dst] += 1
__global__ __launch_bounds__(256) void rgcn_scatter_kernel(
    const float* __restrict__ h, const int* __restrict__ edge_index,
    const int* __restrict__ etype, float* __restrict__ S,
    float* __restrict__ cnt) {
  const int lane = threadIdx.x & 31;
  const long e = (long)blockIdx.x * 8 + (threadIdx.x >> 5);
  if (e >= E_EDGES) return;
  const int src = edge_index[e];
  const int dst = edge_index[(long)E_EDGES + e];
  const int r = etype[e];
  const size_t seg = (size_t)r * N_NODES + (size_t)dst;
  // 32 lanes x float4 = 128 features; h is L2-resident (25.6MB < 192MB L2)
  const float4 v = *(const float4*)(h + (size_t)src * DIM + lane * 4);
  float* p = S + seg * DIM + lane * 4;
  atomicAdd(p + 0, v.x);
  atomicAdd(p + 1, v.y);
  atomicAdd(p + 2, v.z);
  atomicAdd(p + 3, v.w);
  if (lane == 0) atomicAdd(cnt + seg, 1.0f);
}

// -------------------------------------------------- fused GEMM + LN + ReLU
// One wave owns a 16x128 output tile:
//   out[v,:] = sum_r (S[r,v,:]/max(cnt,1)) @ W_r + h[v,:] @ W_root + bias
// then LayerNorm, ReLU, optional residual with h_in.
__global__ __launch_bounds__(128) void rgcn_gemm_ln_kernel(
    const float* __restrict__ hin, const float* __restrict__ S,
    const float* __restrict__ cnt, const float* __restrict__ wrel,
    const float* __restrict__ wroot, const float* __restrict__ bias,
    const float* __restrict__ gmm, const float* __restrict__ bta,
    float* __restrict__ hout, int residual) {
  const int lane = threadIdx.x & 31;
  const int wave = threadIdx.x >> 5;
  const int tile = blockIdx.x * 4 + wave;
  if (tile * 16 >= N_NODES) return;

  const int row = lane & 15;       // A-matrix row / B,C,D column-in-tile
  const int half = lane >> 4;      // 0: K=0,1 / rows 0-7 ; 1: K=2,3 / rows 8-15
  const int koff = half << 1;
  const long rowG = (long)tile * 16 + row;

  v8f acc[8] = {};  // 8 column-tiles of the 16x16 f32 C/D fragment

  for (int mat = 0; mat < 9; ++mat) {
    const float* W;
    const float* Arow;
    float inv;
    if (mat < 8) {
      W = wrel + (size_t)mat * DIM * DIM;
      const size_t seg = (size_t)mat * N_NODES + (size_t)rowG;
      inv = 1.0f / fmaxf(cnt[seg], 1.0f);  // mean = sum * inv
      Arow = S + seg * DIM;
    } else {
      W = wroot;
      inv = 1.0f;
      Arow = hin + (size_t)rowG * DIM;
    }
    for (int k0 = 0; k0 < DIM; k0 += 4) {
      const int k = k0 + koff;
      v2f a;
      a.x = Arow[k] * inv;
      a.y = Arow[k + 1] * inv;
#pragma unroll
      for (int c = 0; c < 8; ++c) {
        const float* Bp = W + (size_t)k * DIM + c * 16 + row;
        v2f b;
        b.x = Bp[0];
        b.y = Bp[DIM];
        acc[c] = __builtin_amdgcn_wmma_f32_16x16x4_f32(
            false, a, false, b, (short)0, acc[c], false, false);
      }
    }
  }

  // ---- epilogue: +bias, LayerNorm over D=128, ReLU, residual, store ----
  float bv[8], gv[8], bev[8];
#pragma unroll
  for (int c = 0; c < 8; ++c) {
    const int col = c * 16 + row;
    bv[c] = bias[col];
    gv[c] = gmm[col];
    bev[c] = bta[col];
  }
#pragma unroll
  for (int m = 0; m < 8; ++m) {  // row (m + 8*half) of the tile
    float xs[8];
    float s = 0.f, sq = 0.f;
#pragma unroll
    for (int c = 0; c < 8; ++c) {
      const float v = acc[c][m] + bv[c];
      xs[c] = v;
      s += v;
      sq += v * v;
    }
    // reduce across the 16 lanes of this half (xor masks 1,2,4,8 stay inside)
#pragma unroll
    for (int off = 1; off < 16; off <<= 1) {
      s += __shfl_xor(s, off, 32);
      sq += __shfl_xor(sq, off, 32);
    }
    const float mu = s * (1.0f / DIM);
    const float var = sq * (1.0f / DIM) - mu * mu;
    const float rs = rsqrtf(var + EPSV);
    const long orow = (long)tile * 16 + m + 8 * half;
#pragma unroll
    for (int c = 0; c < 8; ++c) {
      float y = (xs[c] - mu) * rs * gv[c] + bev[c];
      y = fmaxf(y, 0.0f);
      const size_t idx = (size_t)orow * DIM + (size_t)(c * 16 + row);
      if (residual) y += hin[idx];
      hout[idx] = y;
    }
  }
}

// ---------------------------------------------------------------------------
extern "C" void kernel_launch(void* const* d_in, const int* in_sizes, int n_in,
                              void* d_out, int out_size, void* d_ws,
                              size_t ws_size, hipStream_t stream) {
  (void)in_sizes; (void)n_in; (void)out_size; (void)ws_size;
  const int* x = (const int*)d_in[0];
  const int* edge_index = (const int*)d_in[1];  // [2, E]
  const int* edge_type = (const int*)d_in[2];   // [E]
  const float* emb = (const float*)d_in[3];     // [N, D]
  const float* w_rel = (const float*)d_in[4];   // [L, R, D, D]
  const float* w_root = (const float*)d_in[5];  // [L, D, D]
  const float* bias = (const float*)d_in[6];    // [L, D]
  const float* gmm = (const float*)d_in[7];     // [L, D]
  const float* bta = (const float*)d_in[8];     // [L, D]
  float* out = (float*)d_out;

  // workspace layout (floats)
  float* S = (float*)d_ws;                         // R*N*D  (204.8 MB)
  float* cnt = S + (size_t)RELS * N_NODES * DIM;   // R*N    (1.6 MB)
  float* h0 = cnt + (size_t)RELS * N_NODES;        // N*D    (25.6 MB)
  float* h1 = h0 + (size_t)N_NODES * DIM;          // N*D    (25.6 MB)

  const long zero_f4 = ((long)RELS * N_NODES * DIM + (long)RELS * N_NODES) / 4;
  const int gather_blocks = (int)(((long)N_NODES * (DIM / 4) + 255) / 256);
  const int scatter_blocks = (E_EDGES + 7) / 8;
  const int gemm_blocks = ((N_NODES + 15) / 16 + 3) / 4;

  // h0 = emb[x]
  rgcn_gather_kernel<<<gather_blocks, 256, 0, stream>>>(
      (const float4*)emb, x, (float4*)h0);

  float* hbuf[2] = {h0, h1};
  for (int layer = 0; layer < LAYERS; ++layer) {
    const float* hin = hbuf[layer & 1];
    float* hout = (layer == LAYERS - 1) ? out : hbuf[(layer + 1) & 1];

    rgcn_zero_kernel<<<2048, 256, 0, stream>>>((float4*)S, zero_f4);
    rgcn_scatter_kernel<<<scatter_blocks, 256, 0, stream>>>(
        hin, edge_index, edge_type, S, cnt);
    rgcn_gemm_ln_kernel<<<gemm_blocks, 128, 0, stream>>>(
        hin, S, cnt,
        w_rel + (size_t)layer * RELS * DIM * DIM,
        w_root + (size_t)layer * DIM * DIM,
        bias + (size_t)layer * DIM,
        gmm + (size_t)layer * DIM,
        bta + (size_t)layer * DIM,
        hout, layer > 0 ? 1 : 0);
  }
}